// model_by_2dTo3d_8418135900346
// MI455X (gfx1250) — compile-verified
//
#include <hip/hip_runtime.h>
#include <hip/hip_bf16.h>

typedef __attribute__((ext_vector_type(2))) float v2f;
typedef __attribute__((ext_vector_type(8))) float v8f;

#define D_DEPTH 41
#define CAM_C   64
#define OC      105      // D_DEPTH + CAM_C
#define OCP     112      // padded to 7*16 for WMMA M tiles
#define KDIM    256
#define FH      32
#define FW      88
#define NB      4
#define HW      (FH * FW)           // 2816
#define NPIX    (NB * FH * FW)      // 11264
#define NTILES  (NPIX / 16)         // 704 (exact)
#define NXV     200
#define NYV     200
#define NXC     196                 // cropped X extent of output
#define NPOINT  (NB * D_DEPTH * FH * FW)  // 461824

// ---------------------------------------------------------------------------
// Kernel 1: depth-net GEMM  x[p][o] = sum_k W[o][k]*feat[b][k][h][w] + bias[o]
// fp32 WMMA 16x16x4. One wave per 16-pixel tile; 7 M-tiles (112 rows) per wave
// so the B operand (feat) is loaded once per k-step and reused by 7 WMMAs.
// ---------------------------------------------------------------------------
__global__ __launch_bounds__(128) void k_depthnet_wmma(
    const float* __restrict__ feat, const float* __restrict__ w,
    const float* __restrict__ bias, float* __restrict__ xws)
{
    const int wave = blockIdx.x * (blockDim.x >> 5) + (threadIdx.x >> 5);
    if (wave >= NTILES) return;                       // wave-uniform
    const int lane = threadIdx.x & 31;
    const int col  = lane & 15;                       // N / M row index within tile
    const int hi   = lane >> 4;                       // selects K pair {0,1} vs {2,3}

    const int p   = wave * 16 + col;                  // pixel id (always < NPIX)
    const int b   = p / HW;
    const int rem = p - b * HW;
    const int h   = rem / FW;
    const int wp  = rem - h * FW;
    const float* fp = feat + (size_t)b * KDIM * HW + (size_t)h * FW + wp;

    v8f acc[7] = {};                                  // 7 M-tiles of 16x16 f32

    for (int k0 = 0; k0 < KDIM; k0 += 4) {
        const int kA = k0 + 2 * hi;                   // this half-wave's K pair
        v2f bb;
        bb.x = fp[(size_t)kA * HW];                   // coalesced across lanes 0-15
        bb.y = fp[(size_t)(kA + 1) * HW];
#pragma unroll
        for (int m = 0; m < 7; ++m) {
            const int M  = m * 16 + col;
            const int Mc = (M < OC) ? M : 0;          // clamp: no OOB speculation
            const float* wr = w + (size_t)Mc * KDIM + kA;
            v2f aa;
            aa.x = wr[0];
            aa.y = wr[1];
            if (M >= OC) { aa.x = 0.f; aa.y = 0.f; }  // zero-pad rows 105..111
            acc[m] = __builtin_amdgcn_wmma_f32_16x16x4_f32(
                false, aa, false, bb, (short)0, acc[m], false, false);
        }
    }

    // D tile: vgpr r -> row (r + 8*hi), lane%16 -> column  => pixel-major store
    float* xp = xws + (size_t)p * OCP;
#pragma unroll
    for (int m = 0; m < 7; ++m) {
#pragma unroll
        for (int r = 0; r < 8; ++r) {
            const int o = m * 16 + r + 8 * hi;
            float v = acc[m][r];
            if (o < OC) v += bias[o];                 // pad rows keep garbage, never read
            xp[o] = v;
        }
    }
}

// ---------------------------------------------------------------------------
// Kernel 2: softmax over the 41 depth logits per pixel (in place in xws)
// ---------------------------------------------------------------------------
__global__ __launch_bounds__(256) void k_softmax(float* __restrict__ xws)
{
    const int p = blockIdx.x * blockDim.x + threadIdx.x;
    if (p >= NPIX) return;
    float* xp = xws + (size_t)p * OCP;
    float mx = -3.0e38f;
#pragma unroll
    for (int d = 0; d < D_DEPTH; ++d) mx = fmaxf(mx, xp[d]);
    float e[D_DEPTH];
    float s = 0.f;
#pragma unroll
    for (int d = 0; d < D_DEPTH; ++d) { e[d] = __expf(xp[d] - mx); s += e[d]; }
    const float inv = 1.f / s;
#pragma unroll
    for (int d = 0; d < D_DEPTH; ++d) xp[d] = e[d] * inv;
}

// ---------------------------------------------------------------------------
// Kernel 3: zero the output BEV grid (harness poisons d_out)
// ---------------------------------------------------------------------------
__global__ void k_zero(float* __restrict__ out, int n)
{
    for (int i = blockIdx.x * blockDim.x + threadIdx.x; i < n;
         i += gridDim.x * blockDim.x)
        out[i] = 0.f;
}

// ---------------------------------------------------------------------------
// Kernel 4: fused lift + voxelize + segment-sum.
// One wave per frustum point (b,d,h,w); geometry/kept test is identical across
// the wave -> wave-uniform early exit. Surviving waves: lane c adds
// depth * cam_feat[c] and [c+32] into out[b][c][gx][gy] (crop gx<196 folded in).
// 40MB grid lives in the 192MB L2, so f32 atomics resolve at L2 rate.
// ---------------------------------------------------------------------------
__global__ __launch_bounds__(256) void k_scatter(
    const float* __restrict__ xws, const float* __restrict__ intrins,
    float* __restrict__ out)
{
    const long long t = (long long)blockIdx.x * blockDim.x + threadIdx.x;
    const int  lane = threadIdx.x & 31;
    const int  q    = (int)(t >> 5);                  // point id, uniform per wave
    if (q >= NPOINT) return;

    // q = ((b*41 + d)*FH + h)*FW + w
    const int w  = q % FW;
    const int r1 = q / FW;
    const int h  = r1 % FH;
    const int r2 = r1 / FH;
    const int d  = r2 % D_DEPTH;
    const int b  = r2 / D_DEPTH;

    const int p = (b * FH + h) * FW + w;
    const float* xp = xws + (size_t)p * OCP;
    const float dw = xp[d];                           // softmaxed depth weight

    // 3x3 inverse of intrinsics (cheap VALU, redundant across lanes is fine)
    const float* K = intrins + b * 9;
    const float a00=K[0],a01=K[1],a02=K[2],a10=K[3],a11=K[4],a12=K[5],
                a20=K[6],a21=K[7],a22=K[8];
    const float det = a00*(a11*a22-a12*a21) - a01*(a10*a22-a12*a20)
                    + a02*(a10*a21-a11*a20);
    const float id = 1.f / det;
    const float i00 =  (a11*a22-a12*a21)*id, i01 = -(a01*a22-a02*a21)*id,
                i02 =  (a01*a12-a02*a11)*id;
    const float i10 = -(a10*a22-a12*a20)*id, i11 =  (a00*a22-a02*a20)*id,
                i12 = -(a00*a12-a02*a10)*id;
    const float i20 =  (a10*a21-a11*a20)*id, i21 = -(a00*a21-a01*a20)*id,
                i22 =  (a00*a11-a01*a10)*id;

    const float dv = 4.0f + (float)d;
    const float xs = (float)w * (1407.0f / 87.0f);    // linspace(0,1407,88)
    const float ys = (float)h * (511.0f  / 31.0f);    // linspace(0,511,32)
    const float px = xs * dv, py = ys * dv, pz = dv;

    const float X = i00*px + i01*py + i02*pz;
    const float Y = i10*px + i11*py + i12*pz;
    const float Z = i20*px + i21*py + i22*pz;

    // geom = (int)((pts - (bx - dx/2)) / dx) ; truncation matches astype(int32)
    const int gx = (int)((X + 25.0f) * 4.0f);
    const int gy = (int)( Y          * 4.0f);
    const int gz = (int)((Z + 10.0f) * 0.05f);
    if (gx < 0 || gx >= NXC || gy < 0 || gy >= NYV || gz != 0) return; // uniform

    const int c0 = lane, c1 = lane + 32;
    const float v0 = dw * xp[D_DEPTH + c0];
    const float v1 = dw * xp[D_DEPTH + c1];
    atomicAdd(&out[(((size_t)b * CAM_C + c0) * NXC + gx) * NYV + gy], v0);
    atomicAdd(&out[(((size_t)b * CAM_C + c1) * NXC + gx) * NYV + gy], v1);
}

extern "C" void kernel_launch(void* const* d_in, const int* in_sizes, int n_in,
                              void* d_out, int out_size, void* d_ws, size_t ws_size,
                              hipStream_t stream)
{
    const float* feat    = (const float*)d_in[0];
    const float* intrins = (const float*)d_in[1];
    const float* w_depth = (const float*)d_in[2];
    const float* b_depth = (const float*)d_in[3];
    float* out = (float*)d_out;
    float* xws = (float*)d_ws;   // NPIX * OCP * 4B = 4.81 MB scratch

    k_zero<<<dim3((out_size + 255) / 256), dim3(256), 0, stream>>>(out, out_size);
    k_depthnet_wmma<<<dim3(NTILES / 4), dim3(128), 0, stream>>>(
        feat, w_depth, b_depth, xws);
    k_softmax<<<dim3((NPIX + 255) / 256), dim3(256), 0, stream>>>(xws);
    k_scatter<<<dim3((NPOINT * 32) / 256), dim3(256), 0, stream>>>(
        xws, intrins, out);
}